// GraphSAGE_2869038153785
// MI455X (gfx1250) — compile-verified
//
#include <hip/hip_runtime.h>
#include <math.h>

typedef __attribute__((ext_vector_type(2))) float v2f;
typedef __attribute__((ext_vector_type(8))) float v8f;

#define N_NODES    50000
#define N_EDGES    1250000
#define DIM        64
#define NUM_CLASS  10
#define NUM_GRAPHS 256

// ---------------------------------------------------------------------------
// Grid-stride zero fill (graph-capture safe replacement for hipMemsetAsync).
// ---------------------------------------------------------------------------
__global__ void zero_kernel(float* __restrict__ p, int n) {
  int i = blockIdx.x * blockDim.x + threadIdx.x;
  int stride = gridDim.x * blockDim.x;
  for (; i < n; i += stride) p[i] = 0.0f;
}

// ---------------------------------------------------------------------------
// Edge scatter: agg[dst[e]] += x[src[e]].  16 lanes per edge, float4 per lane
// (GLOBAL_LOAD_B128 gather + 4x GLOBAL_ATOMIC_ADD_F32; both L2-resident).
// ---------------------------------------------------------------------------
__global__ void scatter_add_kernel(const float* __restrict__ x,
                                   const int* __restrict__ src,
                                   const int* __restrict__ dst,
                                   float* __restrict__ agg,
                                   int nEdges) {
  int tid = blockIdx.x * blockDim.x + threadIdx.x;
  int e = tid >> 4;
  if (e >= nEdges) return;
  int part = tid & 15;
  int s = src[e];
  int d = dst[e];
  const float4 v = ((const float4*)(x + (size_t)s * DIM))[part];
  float* o = agg + (size_t)d * DIM + (size_t)part * 4;
  atomicAdd(o + 0, v.x);
  atomicAdd(o + 1, v.y);
  atomicAdd(o + 2, v.z);
  atomicAdd(o + 3, v.w);
}

// ---------------------------------------------------------------------------
// Per-graph sum pool: g[batch[n]] += x[n].
// ---------------------------------------------------------------------------
__global__ void pool_kernel(const float* __restrict__ x,
                            const int* __restrict__ batch,
                            float* __restrict__ g,
                            int nNodes) {
  int tid = blockIdx.x * blockDim.x + threadIdx.x;
  int nd = tid >> 4;
  if (nd >= nNodes) return;
  int part = tid & 15;
  int b = batch[nd];
  const float4 v = ((const float4*)(x + (size_t)nd * DIM))[part];
  float* o = g + (size_t)b * DIM + (size_t)part * 4;
  atomicAdd(o + 0, v.x);
  atomicAdd(o + 1, v.y);
  atomicAdd(o + 2, v.z);
  atomicAdd(o + 3, v.w);
}

// ---------------------------------------------------------------------------
// One K=64 sweep of V_WMMA_F32_16X16X4_F32 accumulation.
//   Arow : base of this lane's A row (row m of the tile), stride-1 in K
//   Wcol : &W[0*64 + col]  (column `col`, stride 64 per K row)
//   kh   : lane>>4 (selects K sub-pair per the 16x4 f32 A layout)
// ---------------------------------------------------------------------------
__device__ inline v8f wmma_k64(const float* __restrict__ Arow,
                               const float* __restrict__ Wcol,
                               int kh, v8f acc) {
#pragma unroll
  for (int k0 = 0; k0 < 64; k0 += 4) {
    int ka = k0 + (kh << 1);
    v2f a;
    a.x = Arow[ka];
    a.y = Arow[ka + 1];
    v2f b;
    b.x = Wcol[(size_t)ka * DIM];
    b.y = Wcol[(size_t)(ka + 1) * DIM];
    acc = __builtin_amdgcn_wmma_f32_16x16x4_f32(
        /*neg_a=*/false, a, /*neg_b=*/false, b,
        /*c_mod=*/(short)0, acc, /*reuse_a=*/false, /*reuse_b=*/false);
  }
  return acc;
}

// ---------------------------------------------------------------------------
// out[rowBase..rowBase+16, :64] = relu( concat(A0,A1)[rows] @ W + bias )
// twoSrc==0 -> K=64 (A0 only, W is 64x64); twoSrc==1 -> K=128 (W is 128x64).
// Block = 128 threads = 4 waves; wave ct owns output columns [ct*16, ct*16+16).
// EXEC is all-ones in every wave (nRows is a multiple of 16), as WMMA requires.
// ---------------------------------------------------------------------------
__global__ void gemm_wmma_kernel(const float* __restrict__ A0,
                                 const float* __restrict__ A1,
                                 const float* __restrict__ W,
                                 const float* __restrict__ bias,
                                 float* __restrict__ out,
                                 int twoSrc) {
  int lane = threadIdx.x & 31;
  int ct   = threadIdx.x >> 5;      // 0..3 -> column tile
  int m    = lane & 15;             // A row / B,D column within tile
  int kh   = lane >> 4;             // K sub-pair select
  int rowBase = blockIdx.x * 16;
  int col  = ct * 16 + m;

  float bv = bias[col];
  v8f acc = {bv, bv, bv, bv, bv, bv, bv, bv};

  const float* Arow0 = A0 + (size_t)(rowBase + m) * DIM;
  const float* Wc0   = W + col;
  acc = wmma_k64(Arow0, Wc0, kh, acc);

  if (twoSrc) {
    const float* Arow1 = A1 + (size_t)(rowBase + m) * DIM;
    const float* Wc1   = W + (size_t)DIM * DIM + col;   // bottom 64 rows of W
    acc = wmma_k64(Arow1, Wc1, kh, acc);
  }

#pragma unroll
  for (int j = 0; j < 8; ++j) {
    int r = rowBase + kh * 8 + j;   // D layout: VGPR j -> row j (+8 for kh=1)
    float v = fmaxf(acc[j], 0.0f);  // ReLU fused
    out[(size_t)r * DIM + col] = v;
  }
}

// ---------------------------------------------------------------------------
// logits = t @ Wd2 + bd2 ; softmax rows.  One thread per graph (256 total).
// ---------------------------------------------------------------------------
__global__ void head_kernel(const float* __restrict__ t,
                            const float* __restrict__ Wd2,
                            const float* __restrict__ bd2,
                            float* __restrict__ out) {
  int r = blockIdx.x * blockDim.x + threadIdx.x;
  if (r >= NUM_GRAPHS) return;
  float logits[NUM_CLASS];
#pragma unroll
  for (int c = 0; c < NUM_CLASS; ++c) logits[c] = bd2[c];
  const float* tr = t + (size_t)r * DIM;
  for (int k = 0; k < DIM; ++k) {
    float tv = tr[k];
#pragma unroll
    for (int c = 0; c < NUM_CLASS; ++c)
      logits[c] = fmaf(tv, Wd2[k * NUM_CLASS + c], logits[c]);
  }
  float mx = logits[0];
#pragma unroll
  for (int c = 1; c < NUM_CLASS; ++c) mx = fmaxf(mx, logits[c]);
  float sum = 0.0f;
#pragma unroll
  for (int c = 0; c < NUM_CLASS; ++c) {
    float e = __expf(logits[c] - mx);
    logits[c] = e;
    sum += e;
  }
  float inv = 1.0f / sum;
#pragma unroll
  for (int c = 0; c < NUM_CLASS; ++c)
    out[(size_t)r * NUM_CLASS + c] = logits[c] * inv;
}

// ---------------------------------------------------------------------------
extern "C" void kernel_launch(void* const* d_in, const int* in_sizes, int n_in,
                              void* d_out, int out_size, void* d_ws, size_t ws_size,
                              hipStream_t stream) {
  const float* x0   = (const float*)d_in[0];   // node_embeddings [N,64]
  const int*   edge = (const int*)  d_in[1];   // edge_index [2,E]
  const int*   batch= (const int*)  d_in[2];   // [N]
  const float* W1   = (const float*)d_in[3];   // [128,64]
  const float* b1   = (const float*)d_in[4];
  const float* W2   = (const float*)d_in[5];   // [128,64]
  const float* b2   = (const float*)d_in[6];
  const float* Wd1  = (const float*)d_in[7];   // [64,64]
  const float* bd1  = (const float*)d_in[8];
  const float* Wd2  = (const float*)d_in[9];   // [64,10]
  const float* bd2  = (const float*)d_in[10];
  float* out = (float*)d_out;

  const int* src = edge;            // edge_index[0]
  const int* dst = edge + N_EDGES;  // edge_index[1]

  // Workspace carve-out (~38.6 MB).
  float* agg = (float*)d_ws;                          // [N,64]
  float* x1  = agg + (size_t)N_NODES * DIM;           // [N,64]
  float* x2  = x1  + (size_t)N_NODES * DIM;           // [N,64]
  float* g   = x2  + (size_t)N_NODES * DIM;           // [256,64]
  float* t   = g   + (size_t)NUM_GRAPHS * DIM;        // [256,64]

  const int nAgg = N_NODES * DIM;                     // 3.2M floats
  const int scatterBlocks = (N_EDGES * 16 + 255) / 256;   // 78125
  const int poolBlocks    = (N_NODES * 16 + 255) / 256;   // 3125

  // ---- Layer 1 ----
  zero_kernel<<<2048, 256, 0, stream>>>(agg, nAgg);
  scatter_add_kernel<<<scatterBlocks, 256, 0, stream>>>(x0, src, dst, agg, N_EDGES);
  gemm_wmma_kernel<<<N_NODES / 16, 128, 0, stream>>>(x0, agg, W1, b1, x1, 1);

  // ---- Layer 2 ----
  zero_kernel<<<2048, 256, 0, stream>>>(agg, nAgg);
  scatter_add_kernel<<<scatterBlocks, 256, 0, stream>>>(x1, src, dst, agg, N_EDGES);
  gemm_wmma_kernel<<<N_NODES / 16, 128, 0, stream>>>(x1, agg, W2, b2, x2, 1);

  // ---- Graph pooling ----
  zero_kernel<<<64, 256, 0, stream>>>(g, NUM_GRAPHS * DIM);
  pool_kernel<<<poolBlocks, 256, 0, stream>>>(x2, batch, g, N_NODES);

  // ---- Dense head ----
  gemm_wmma_kernel<<<NUM_GRAPHS / 16, 128, 0, stream>>>(g, nullptr, Wd1, bd1, t, 0);
  head_kernel<<<1, 256, 0, stream>>>(t, Wd2, bd2, out);
}